// WhisperDecoderAttention_4681514353079
// MI455X (gfx1250) — compile-verified
//
#include <hip/hip_runtime.h>

typedef float v2f __attribute__((ext_vector_type(2)));
typedef float v8f __attribute__((ext_vector_type(8)));
typedef unsigned int v4u __attribute__((ext_vector_type(4)));
typedef unsigned int v8u __attribute__((ext_vector_type(8)));

#define NB    16
#define NH    20
#define NS    8
#define HD    64
#define HH    1280
#define PAST  4096
#define TTOT  4104      // PAST + NS
#define TPAD  4112      // padded score row stride (multiple of 16)
#define NTILES 257      // ceil(TTOT/16)

static __device__ __forceinline__ v8f wmma4(v2f a, v2f b, v8f c) {
  // D = A(16x4 f32) * B(4x16 f32) + C(16x16 f32)
  return __builtin_amdgcn_wmma_f32_16x16x4_f32(false, a, false, b, (short)0, c, false, false);
}

// ---------------------------------------------------------------------------
// Kernel 1: Q/K/V projections. 128x1280 @ 1280x1280 per matrix, WMMA 16x16 tiles.
// Q -> q_ws in [b,h,s,d]; curr K/V -> rows PAST..PAST+7 of key_out/value_out.
// grid: 240 blocks x 256 threads = 1920 waves = 3 * (8 Mtiles) * (80 Ntiles)
// ---------------------------------------------------------------------------
__global__ void __launch_bounds__(256) proj_qkv_kernel(
    const float* __restrict__ hs,
    const float* __restrict__ Wq, const float* __restrict__ bq,
    const float* __restrict__ Wk,
    const float* __restrict__ Wv, const float* __restrict__ bv,
    float* __restrict__ q_ws, float* __restrict__ key_out, float* __restrict__ value_out) {
  int g    = blockIdx.x * 8 + (threadIdx.x >> 5);
  int lane = threadIdx.x & 31;
  int half = lane >> 4, lr = lane & 15;
  int w    = g / 640;           // 0=Q 1=K 2=V
  int rem  = g % 640;
  int mt   = rem / 80, nt = rem % 80;
  const float* W = (w == 0) ? Wq : (w == 1) ? Wk : Wv;
  int col = nt * 16 + lr;
  const float* arow = hs + (size_t)(mt * 16 + lr) * HH + 2 * half;

  v8f acc = {0,0,0,0,0,0,0,0};
  for (int k0 = 0; k0 < HH; k0 += 4) {
    v2f a = *(const v2f*)(arow + k0);                       // A[M=lr][K=k0+2h .. +1]
    v2f b;
    b.x = W[(size_t)(k0 + 2*half)     * HH + col];          // B[K][N=col]
    b.y = W[(size_t)(k0 + 2*half + 1) * HH + col];
    acc = wmma4(a, b, acc);
  }
#pragma unroll
  for (int v = 0; v < 8; ++v) {
    int r  = mt * 16 + v + 8 * half;   // C: vgpr v, upper half-wave = M+8
    int b_ = r >> 3, s = r & 7;
    int h  = col >> 6, d = col & 63;
    if (w == 0) {
      q_ws[(((size_t)b_ * NH + h) * NS + s) * HD + d] = acc[v] + bq[col];
    } else if (w == 1) {
      key_out[(((size_t)b_ * NH + h) * TTOT + PAST + s) * HD + d] = acc[v];
    } else {
      value_out[(((size_t)b_ * NH + h) * TTOT + PAST + s) * HD + d] = acc[v] + bv[col];
    }
  }
}

// ---------------------------------------------------------------------------
// Kernel 2: fused KV-cache key copy + Q*K^T scores.
// One wave per (b,h,16-key tile). For past tiles, the Tensor Data Mover loads
// the 16x64 f32 tile directly into LDS with hardware padding (pad every 64
// DWORDs by 4 DWORDs -> stride-68 rows, bank-conflict-free B-fragment reads),
// then the wave copies LDS -> key_out (cache concat) and runs 16 chained
// WMMA f32 16x16x4 over head_dim=64. All descriptor fields are blockIdx-
// derived, i.e. wave-uniform -> legal SGPR operands.
// grid: 16*20*257 blocks x 32 threads.
// ---------------------------------------------------------------------------
__global__ void __launch_bounds__(32) scores_key_kernel(
    const float* __restrict__ past_key, const float* __restrict__ q_ws,
    float* __restrict__ key_out, float* __restrict__ scores) {
  __shared__ float skey[16 * 68];
  __shared__ float sq[16 * 68];
  int bh   = blockIdx.x / NTILES;
  int tile = blockIdx.x % NTILES;
  int lane = threadIdx.x;
  int half = lane >> 4, lr = lane & 15;
  float* ko = key_out + (size_t)bh * TTOT * HD;

  if (tile < 256) {
    // ---- TDM: past_key 16x64 tile -> LDS (padded) ----
    unsigned long long ga =
        (unsigned long long)(size_t)(past_key + (size_t)bh * PAST * HD + (size_t)tile * 16 * HD);
    unsigned lds_off = (unsigned)(size_t)(void*)skey;   // generic low 32 bits = LDS byte offset
    v4u g0;
    g0[0] = 1u;                                   // count=1, user descriptor
    g0[1] = lds_off;                              // lds_addr
    g0[2] = (unsigned)ga;                         // global_addr[31:0]
    g0[3] = (unsigned)((ga >> 32) & 0x1FFFFFFu)   // global_addr[56:32]
          | 0x80000000u;                          // type = 2 ("image")
    v8u g1;
    g1[0] = (2u << 16)        // data_size = 4 bytes
          | (1u << 20)        // pad_enable
          | (5u << 22)        // pad_interval: 64 DWORDs
          | (3u << 25);       // pad_amount:   4 DWORDs  -> row stride 68 floats
    g1[1] = (64u << 16);      // tensor_dim0 = 64 (dim0[15:0] in bits 63:48)
    g1[2] = (16u << 16);      // tensor_dim1 = 16 (dim1[15:0] in bits 95:80)
    g1[3] = (64u << 16);      // tile_dim0 = 64   (bits 127:112)
    g1[4] = 16u;              // tile_dim1 = 16   (bits 143:128)
    g1[5] = 64u;              // tensor_dim0_stride = 64 (bits 191:160)
    g1[6] = 0u;               // dim0_stride hi / dim1_stride lo
    g1[7] = 0u;               // tensor_dim1_stride (unused for 2D tile)
    asm volatile("tensor_load_to_lds %0, %1" :: "s"(g0), "s"(g1) : "memory");
    asm volatile("s_wait_tensorcnt 0x0" ::: "memory");
    // cache concat: LDS tile -> key_out (past_key read from HBM exactly once)
    for (int i = 0; i < 8; ++i) {
      int idx = lane + 32 * i;
      int row = idx >> 4, c4 = (idx & 15) * 4;
      float4 val = *(const float4*)(&skey[row * 68 + c4]);
      *(float4*)(ko + (size_t)(tile * 16 + row) * HD + c4) = val;
    }
  } else {
    // tail tile: rows 0..7 are curr_k (already in key_out), rows 8..15 zero
    for (int i = 0; i < 8; ++i) {
      int idx = lane + 32 * i;
      int row = idx >> 4, c4 = (idx & 15) * 4;
      float4 val = make_float4(0.f, 0.f, 0.f, 0.f);
      if (row < 8) val = *(const float4*)(ko + (size_t)(PAST + row) * HD + c4);
      *(float4*)(&skey[row * 68 + c4]) = val;
    }
  }
  // stage 16x64 q tile (rows 8..15 zero)
  const float* qb = q_ws + (size_t)bh * NS * HD;
  for (int i = 0; i < 8; ++i) {
    int idx = lane + 32 * i;
    int row = idx >> 4, c4 = (idx & 15) * 4;
    float4 val = make_float4(0.f, 0.f, 0.f, 0.f);
    if (row < 8) val = *(const float4*)(qb + row * HD + c4);
    *(float4*)(&sq[row * 68 + c4]) = val;
  }
  __syncthreads();

  v8f acc = {0,0,0,0,0,0,0,0};
#pragma unroll
  for (int kk = 0; kk < 16; ++kk) {
    int off = 4 * kk + 2 * half;
    v2f a = *(const v2f*)(&sq[lr * 68 + off]);     // A[M=lr][K..]
    v2f b = *(const v2f*)(&skey[lr * 68 + off]);   // B[K..][N=lr] = key[row lr][K..]
    acc = wmma4(a, b, acc);
  }
  if (half == 0) {                                  // rows 0..7 live in lower half-wave
    int t = tile * 16 + lr;
    if (t < TTOT) {
#pragma unroll
      for (int v = 0; v < 8; ++v)
        scores[((size_t)bh * NS + v) * TPAD + t] = acc[v];
    }
  }
}

// ---------------------------------------------------------------------------
// Kernel 3: softmax over T per (b,h,s) row (applies 1/sqrt(64) inside exp),
// zeroes the TPAD padding so kernel 4 can read full 16-wide tiles.
// ---------------------------------------------------------------------------
__global__ void __launch_bounds__(256) softmax_kernel(float* __restrict__ scores) {
  __shared__ float red[256];
  int tid  = threadIdx.x;
  float* p = scores + (size_t)blockIdx.x * TPAD;
  float m = -1e30f;
  for (int i = tid; i < TTOT; i += 256) m = fmaxf(m, p[i]);
  red[tid] = m; __syncthreads();
  for (int s = 128; s > 0; s >>= 1) { if (tid < s) red[tid] = fmaxf(red[tid], red[tid + s]); __syncthreads(); }
  m = red[0];
  __syncthreads();
  float sum = 0.f;
  for (int i = tid; i < TTOT; i += 256) { float e = __expf((p[i] - m) * 0.125f); p[i] = e; sum += e; }
  red[tid] = sum; __syncthreads();
  for (int s = 128; s > 0; s >>= 1) { if (tid < s) red[tid] += red[tid + s]; __syncthreads(); }
  float inv = 1.0f / red[0];
  for (int i = tid; i < TTOT; i += 256) p[i] *= inv;
  for (int i = TTOT + tid; i < TPAD; i += 256) p[i] = 0.f;
}

// ---------------------------------------------------------------------------
// Kernel 4: fused value-cache copy + ctx = P * V.
// One WG per (b,h); 8 waves stride the 257 key tiles, each accumulating four
// 16x16 WMMA C-tiles (N=64), combined via LDS ds_add_f32.
// ---------------------------------------------------------------------------
__global__ void __launch_bounds__(256) ctx_value_kernel(
    const float* __restrict__ past_value, const float* __restrict__ probs,
    float* __restrict__ value_out, float* __restrict__ ctx_ws) {
  __shared__ float sval[8][16 * 68];
  __shared__ float sctx[16 * 64];
  int tid  = threadIdx.x;
  int wave = tid >> 5, lane = tid & 31;
  int half = lane >> 4, lr = lane & 15;
  int bh   = blockIdx.x;
  for (int i = tid; i < 16 * 64; i += 256) sctx[i] = 0.f;
  __syncthreads();

  const float* pv = past_value + (size_t)bh * PAST * HD;
  float*       vo = value_out  + (size_t)bh * TTOT * HD;
  const float* pr = probs + ((size_t)bh * NS + (lr < 8 ? lr : 0)) * TPAD;
  float* my = sval[wave];

  v8f acc[4];
#pragma unroll
  for (int nt = 0; nt < 4; ++nt) acc[nt] = (v8f){0,0,0,0,0,0,0,0};

  for (int tile = wave; tile < NTILES; tile += 8) {
    // stage 16x64 value tile (copy past -> value_out along the way)
    for (int i = 0; i < 8; ++i) {
      int idx = lane + 32 * i;
      int row = idx >> 4, c4 = (idx & 15) * 4;
      float4 val;
      if (tile < 256) {
        int t = tile * 16 + row;
        val = *(const float4*)(pv + (size_t)t * HD + c4);
        *(float4*)(vo + (size_t)t * HD + c4) = val;         // cache concat
      } else {
        if (row < 8) val = *(const float4*)(vo + (size_t)(PAST + row) * HD + c4);
        else         val = make_float4(0.f, 0.f, 0.f, 0.f);
      }
      *(float4*)(&my[row * 68 + c4]) = val;
    }
    __threadfence_block();   // order per-wave LDS store -> LDS read (no barrier: wave-private region)

    v2f af[4];
#pragma unroll
    for (int kk = 0; kk < 4; ++kk) {
      if (lr < 8) af[kk] = *(const v2f*)(pr + tile * 16 + 4 * kk + 2 * half);
      else        { af[kk].x = 0.f; af[kk].y = 0.f; }       // pad query rows 8..15
    }
#pragma unroll
    for (int nt = 0; nt < 4; ++nt) {
#pragma unroll
      for (int kk = 0; kk < 4; ++kk) {
        v2f b;                                              // B[K][N] = value[t0+K][nt*16+lr]
        b.x = my[(4 * kk + 2 * half)     * 68 + nt * 16 + lr];
        b.y = my[(4 * kk + 2 * half + 1) * 68 + nt * 16 + lr];
        acc[nt] = wmma4(af[kk], b, acc[nt]);
      }
    }
  }
  // combine partial ctx across waves (rows 8..15 are exact zeros)
#pragma unroll
  for (int nt = 0; nt < 4; ++nt)
#pragma unroll
    for (int v = 0; v < 8; ++v)
      atomicAdd(&sctx[(v + 8 * half) * 64 + nt * 16 + lr], acc[nt][v]);
  __syncthreads();

  int b_ = bh / NH, h = bh % NH;
  for (int j = tid; j < NS * HD; j += 256) {
    int s = j >> 6, d = j & 63;
    ctx_ws[((size_t)b_ * NS + s) * HH + h * 64 + d] = sctx[s * 64 + d];  // merge heads
  }
}

// ---------------------------------------------------------------------------
// Kernel 5: out = ctx @ Wd + bd   (128x1280 @ 1280x1280)
// ---------------------------------------------------------------------------
__global__ void __launch_bounds__(256) out_proj_kernel(
    const float* __restrict__ ctx, const float* __restrict__ Wd,
    const float* __restrict__ bd, float* __restrict__ out) {
  int g    = blockIdx.x * 8 + (threadIdx.x >> 5);
  int lane = threadIdx.x & 31;
  int half = lane >> 4, lr = lane & 15;
  int mt   = g / 80, nt = g % 80;
  int col  = nt * 16 + lr;
  const float* arow = ctx + (size_t)(mt * 16 + lr) * HH + 2 * half;

  v8f acc = {0,0,0,0,0,0,0,0};
  for (int k0 = 0; k0 < HH; k0 += 4) {
    v2f a = *(const v2f*)(arow + k0);
    v2f b;
    b.x = Wd[(size_t)(k0 + 2*half)     * HH + col];
    b.y = Wd[(size_t)(k0 + 2*half + 1) * HH + col];
    acc = wmma4(a, b, acc);
  }
#pragma unroll
  for (int v = 0; v < 8; ++v) {
    int r = mt * 16 + v + 8 * half;
    out[(size_t)r * HH + col] = acc[v] + bd[col];
  }
}

// ---------------------------------------------------------------------------
extern "C" void kernel_launch(void* const* d_in, const int* in_sizes, int n_in,
                              void* d_out, int out_size, void* d_ws, size_t ws_size,
                              hipStream_t stream) {
  const float* hs         = (const float*)d_in[0];
  const float* past_key   = (const float*)d_in[1];
  const float* past_value = (const float*)d_in[2];
  // d_in[3] = cache_mask (int32): cache_length == PAST statically, unused.
  const float* Wq = (const float*)d_in[4];
  const float* bq = (const float*)d_in[5];
  const float* Wk = (const float*)d_in[6];
  const float* Wv = (const float*)d_in[7];
  const float* bv = (const float*)d_in[8];
  const float* Wd = (const float*)d_in[9];
  const float* bd = (const float*)d_in[10];

  float* out       = (float*)d_out;                         // [16,8,1280]
  float* key_out   = out + (size_t)NB * NS * HH;            // [16,20,4104,64]
  float* value_out = key_out + (size_t)NB * NH * TTOT * HD; // [16,20,4104,64]

  float* q_ws   = (float*)d_ws;                             // [16,20,8,64]
  float* scores = q_ws + (size_t)NB * NH * NS * HD;         // [16*20*8, 4112]
  float* ctx_ws = scores + (size_t)NB * NH * NS * TPAD;     // [16,8,1280]  (~43.4 MB total)

  proj_qkv_kernel <<<240,            256, 0, stream>>>(hs, Wq, bq, Wk, Wv, bv, q_ws, key_out, value_out);
  scores_key_kernel<<<NB*NH*NTILES,  32,  0, stream>>>(past_key, q_ws, key_out, scores);
  softmax_kernel  <<<NB*NH*NS,       256, 0, stream>>>(scores);
  ctx_value_kernel<<<NB*NH,          256, 0, stream>>>(past_value, scores, value_out, ctx_ws);
  out_proj_kernel <<<80,             256, 0, stream>>>(ctx_ws, Wd, bd, out);
}